// InferSent_10196252360952
// MI455X (gfx1250) — compile-verified
//
#include <hip/hip_runtime.h>
#include <hip/hip_bf16.h>
#include <math.h>

typedef __attribute__((ext_vector_type(2))) float v2f;
typedef __attribute__((ext_vector_type(8))) float v8f;
typedef __attribute__((ext_vector_type(4))) unsigned u32x4;
typedef __attribute__((ext_vector_type(8))) unsigned u32x8;

// ---------------- problem constants ----------------
#define T_LEN 2048
#define P_LEN 256
#define D_IN  300
#define H1    600
#define G1    2400   // 4*H1
#define H2    2048
#define G2    8192   // 4*H2
#define E3    6144   // 3*H2

// ---------------- workspace layout (float offsets) ----------------
#define OFF_SYNC  0            // 16 u32 (lstm1 uses [0..1], lstm2 uses [4..5])
#define OFF_HB1   16           // 2*640 double-buffered h for LSTM1
#define OFF_HB2   1296         // 2*2048 double-buffered h for LSTM2
#define ZERO_FLOATS 5392
#define OFF_PRE1  5392                         // 2048*2400
#define OFF_HALL  (OFF_PRE1 + 4915200)         // 2048*600
#define OFF_EMBG  (OFF_HALL + 1228800)         // 256*600
#define OFF_EMB2  (OFF_EMBG + 153600)          // 256*600
#define OFF_PRE2  (OFF_EMB2 + 153600)          // 256*8192
#define OFF_PH    (OFF_PRE2 + 2097152)         // 256*2048
#define OFF_S     (OFF_PH   + 524288)          // 256*2048
#define OFF_QKV   (OFF_S    + 524288)          // 256*6144
#define OFF_SCOR  (OFF_QKV  + 1572864)         // 256*256
#define OFF_AOUT  (OFF_SCOR + 65536)           // 256*2048
#define OFF_CSUM  (OFF_AOUT + 524288)          // 2048

// ---------------- grid-wide barrier (agent-scope acq/rel atomics) ----------------
__device__ __forceinline__ void grid_barrier(unsigned* sync, unsigned nwg, unsigned* gen) {
  __syncthreads();
  if (threadIdx.x == 0) {
    unsigned target = *gen + 1u;
    unsigned arrived = __hip_atomic_fetch_add(&sync[0], 1u, __ATOMIC_ACQ_REL,
                                              __HIP_MEMORY_SCOPE_AGENT);
    if (arrived == nwg - 1u) {
      __hip_atomic_store(&sync[0], 0u, __ATOMIC_RELAXED, __HIP_MEMORY_SCOPE_AGENT);
      __hip_atomic_store(&sync[1], target, __ATOMIC_RELEASE, __HIP_MEMORY_SCOPE_AGENT);
    } else {
      while (__hip_atomic_load(&sync[1], __ATOMIC_ACQUIRE, __HIP_MEMORY_SCOPE_AGENT) < target)
        __builtin_amdgcn_s_sleep(1);
    }
    *gen = target;
  }
  __syncthreads();
}

__device__ __forceinline__ float sigmoidf_(float x) { return 1.f / (1.f + expf(-x)); }

// ---------------- generic 16x16-tile WMMA f32 GEMM ----------------
// C[M,N] = act( scale * (A @ op(B)) + bias0 + bias1 )
// TRANSB: B is (N,K) row-major (compute A@B^T).  else B is (K,N) row-major.
template <bool TRANSB, int ACT>
__global__ void gemm16(const float* __restrict__ A, const float* __restrict__ B,
                       const float* __restrict__ bias0, const float* __restrict__ bias1,
                       float* __restrict__ C, int M, int N, int K,
                       int lda, int ldb, int ldc, float scale) {
  (void)M;
  const int lane = threadIdx.x;          // one wave32 per 16x16 tile
  const int m0 = blockIdx.y * 16;
  const int n0 = blockIdx.x * 16;
  const int r  = lane & 15;
  const int klo = (lane >> 4) << 1;      // 0 or 2 (documented 16x4 f32 A layout)

  v8f acc = {};
  const float* arow = A + (size_t)(m0 + r) * lda + klo;

  if (TRANSB) {
    const int bn = n0 + r;
    const bool bvalid = bn < N;
    const float* brow = B + (size_t)(bvalid ? bn : 0) * ldb + klo;
    for (int k = 0; k < K; k += 4) {
      v2f a; a.x = arow[k]; a.y = arow[k + 1];
      v2f b;
      float bx = brow[k], by = brow[k + 1];
      b.x = bvalid ? bx : 0.f;
      b.y = bvalid ? by : 0.f;
      acc = __builtin_amdgcn_wmma_f32_16x16x4_f32(false, a, false, b, (short)0, acc,
                                                  false, false);
    }
  } else {
    const float* bcol = B + (size_t)klo * ldb + (n0 + r);
    for (int k = 0; k < K; k += 4) {
      v2f a; a.x = arow[k]; a.y = arow[k + 1];
      v2f b; b.x = bcol[(size_t)k * ldb]; b.y = bcol[(size_t)(k + 1) * ldb];
      acc = __builtin_amdgcn_wmma_f32_16x16x4_f32(false, a, false, b, (short)0, acc,
                                                  false, false);
    }
  }

  const int n = n0 + r;
  const int mbase = m0 + ((lane >> 4) << 3);
  if (n < N) {
    float bv = 0.f;
    if (bias0) bv += bias0[n];
    if (bias1) bv += bias1[n];
#pragma unroll
    for (int i = 0; i < 8; ++i) {
      float v = acc[i] * scale + bv;
      if (ACT == 1) v = tanhf(v);
      C[(size_t)(mbase + i) * ldc + n] = v;
    }
  }
}

// ---------------- persistent LSTM1 scan: 40 WGs x 15 hidden units ----------------
// h staged into LDS with GLOBAL_LOAD_ASYNC_TO_LDS_B128 (ASYNCcnt path).
__global__ void lstm1_scan(const float* __restrict__ pre, const float* __restrict__ Whh,
                           const float* __restrict__ action, float* __restrict__ hbuf,
                           float* __restrict__ hall, unsigned* sync) {
  __shared__ __align__(16) float hlds[608];
  __shared__ float gl[64];
  const int tid = threadIdx.x, wg = blockIdx.x;
  const int gate = tid & 3;
  const int j = gate * H1 + (wg * 15 + (tid >> 2));     // gate row, valid tid<60
  const float4* wrow = (const float4*)(Whh + (size_t)(tid < 60 ? j : 0) * H1);
  const unsigned lds0 = (unsigned)(size_t)&hlds[0];     // LDS byte offset (low 32 bits)
  float creg = 0.f;
  unsigned gen = 0;

  for (int t = 0; t < T_LEN; ++t) {
    const float* hc = hbuf + (t & 1) * 640;
    // async copy 2400 bytes: 64 lanes x 16B strips at offsets 0 / 1024, tail 22 lanes
    {
      unsigned ldsa = lds0 + (unsigned)tid * 16u;
      unsigned goff = (unsigned)tid * 16u;
      asm volatile("global_load_async_to_lds_b128 %0, %1, %2"
                   :: "v"(ldsa), "v"(goff), "s"(hc) : "memory");
      asm volatile("global_load_async_to_lds_b128 %0, %1, %2 offset:1024"
                   :: "v"(ldsa), "v"(goff), "s"(hc) : "memory");
      if (tid < 22)
        asm volatile("global_load_async_to_lds_b128 %0, %1, %2 offset:2048"
                     :: "v"(ldsa), "v"(goff), "s"(hc) : "memory");
      asm volatile("s_wait_asynccnt 0x0" ::: "memory");
    }
    __syncthreads();
    if (tid < 60) {
      float acc = pre[(size_t)t * G1 + j];
      const float4* hv = (const float4*)hlds;
#pragma unroll 4
      for (int k = 0; k < H1 / 4; ++k) {
        float4 w = wrow[k], h4 = hv[k];
        acc += w.x * h4.x + w.y * h4.y + w.z * h4.z + w.w * h4.w;
      }
      gl[tid] = acc;
    }
    __syncthreads();
    if (tid < 15) {
      float i_s = sigmoidf_(gl[tid * 4 + 0]);
      float f_s = sigmoidf_(gl[tid * 4 + 1]);
      float gg  = gl[tid * 4 + 2];
      float o_s = sigmoidf_(gl[tid * 4 + 3]);
      float cn = f_s * creg + i_s * tanhf(gg);
      float hn = o_s * tanhf(cn);
      float m = 1.f - action[t];
      int u = wg * 15 + tid;
      hall[(size_t)t * H1 + u] = hn;                 // unmasked output
      hbuf[((t + 1) & 1) * 640 + u] = hn * m;        // masked carry (double buffer)
      creg = cn * m;
    }
    grid_barrier(sync, 40, &gen);
  }
}

// ---------------- persistent LSTM2 scan: 64 WGs x 32 hidden units ----------------
// h staged into LDS by the Tensor Data Mover (one 1x2048 f32 tile per step, wave 0).
__global__ void lstm2_scan(const float* __restrict__ pre, const float* __restrict__ Whh,
                           float* __restrict__ hbuf, float* __restrict__ ph,
                           unsigned* sync) {
  __shared__ __align__(16) float hlds[H2];
  __shared__ float gl[128];
  const int tid = threadIdx.x, wg = blockIdx.x;
  const int gate = tid & 3;
  const int j = gate * H2 + (wg * 32 + (tid >> 2));     // 128 rows, all threads active
  const float4* wrow = (const float4*)(Whh + (size_t)j * H2);
  const unsigned lds0 = (unsigned)(size_t)&hlds[0];
  float creg = 0.f;
  unsigned gen = 0;

  for (int t = 0; t < P_LEN; ++t) {
    if (tid < 32) {                                     // wave 0 issues the TDM DMA
      unsigned long long ga = (unsigned long long)(size_t)(hbuf + (t & 1) * H2);
      u32x4 g0;
      g0[0] = 1u;                                       // count=1, no gather
      g0[1] = lds0;                                     // lds_addr (bytes)
      g0[2] = (unsigned)ga;                             // global_addr[31:0]
      g0[3] = (unsigned)((ga >> 32) & 0x1ffffffull)     // global_addr[56:32]
              | (2u << 30);                             // type=2 (image)
      u32x8 g1;
      g1[0] = 2u << 16;                                 // data_size=4B, wg_mask=0
      g1[1] = ((unsigned)H2 & 0xffffu) << 16;           // tensor_dim0 lo16
      g1[2] = ((unsigned)H2 >> 16) | (1u << 16);        // tensor_dim0 hi16 | tensor_dim1 lo16
      g1[3] = ((unsigned)H2) << 16;                     // tensor_dim1 hi16(0) | tile_dim0
      g1[4] = 1u;                                       // tile_dim1=1, tile_dim2=0
      g1[5] = (unsigned)H2;                             // tensor_dim0_stride lo32
      g1[6] = 0u;
      g1[7] = 0u;
      asm volatile("tensor_load_to_lds %0, %1\n\t"
                   "s_wait_tensorcnt 0x0"
                   :: "s"(g0), "s"(g1) : "memory");
    }
    __syncthreads();
    if (t + 1 < P_LEN) __builtin_prefetch(&pre[(size_t)(t + 1) * G2 + j], 0, 1);
    {
      float acc = pre[(size_t)t * G2 + j];
      const float4* hv = (const float4*)hlds;
#pragma unroll 4
      for (int k = 0; k < H2 / 4; ++k) {
        float4 w = wrow[k], h4 = hv[k];
        acc += w.x * h4.x + w.y * h4.y + w.z * h4.z + w.w * h4.w;
      }
      gl[tid] = acc;
    }
    __syncthreads();
    if (tid < 32) {
      float i_s = sigmoidf_(gl[tid * 4 + 0]);
      float f_s = sigmoidf_(gl[tid * 4 + 1]);
      float gg  = gl[tid * 4 + 2];
      float o_s = sigmoidf_(gl[tid * 4 + 3]);
      float cn = f_s * creg + i_s * tanhf(gg);
      float hn = o_s * tanhf(cn);
      int u = wg * 32 + tid;
      ph[(size_t)t * H2 + u] = hn;
      hbuf[((t + 1) & 1) * H2 + u] = hn;             // zero action => no masking
      creg = cn;
    }
    grid_barrier(sync, 64, &gen);
  }
}

// ---------------- helpers ----------------
__global__ void gather_rows(const float* __restrict__ hall, const int* __restrict__ pos,
                            float* __restrict__ out) {
  const int p = blockIdx.x;
  const int src = pos[p];
  for (int i = threadIdx.x; i < H1; i += blockDim.x)
    out[(size_t)p * H1 + i] = hall[(size_t)src * H1 + i];
}

__global__ void softmax256(float* __restrict__ sc) {
  __shared__ float red[256];
  const int l = blockIdx.x, t = threadIdx.x;
  float v = sc[(size_t)l * 256 + t];
  red[t] = v; __syncthreads();
  for (int s = 128; s > 0; s >>= 1) { if (t < s) red[t] = fmaxf(red[t], red[t + s]); __syncthreads(); }
  float mx = red[0]; __syncthreads();
  float e = expf(v - mx);
  red[t] = e; __syncthreads();
  for (int s = 128; s > 0; s >>= 1) { if (t < s) red[t] += red[t + s]; __syncthreads(); }
  sc[(size_t)l * 256 + t] = e / red[0];
}

__global__ void colsum_k(const float* __restrict__ aout, float* __restrict__ cs) {
  const int n = blockIdx.x * 256 + threadIdx.x;
  float s = 0.f;
  for (int l = 0; l < P_LEN; ++l) s += aout[(size_t)l * H2 + n];
  cs[n] = s;
}

// out[n] = dot(csum, Wo[n,:]) + 256*bo[n]   (sum over L folded by linearity)
__global__ void final_matvec(const float* __restrict__ cs, const float* __restrict__ Wo,
                             const float* __restrict__ bo, float* __restrict__ out) {
  const int wid = threadIdx.x >> 5;
  const int lane = threadIdx.x & 31;
  const int n = blockIdx.x * 8 + wid;     // 256 blocks * 8 waves = 2048 outputs
  const float* wrow = Wo + (size_t)n * H2;
  float acc = 0.f;
  for (int k = lane; k < H2; k += 32) acc += cs[k] * wrow[k];
  for (int off = 16; off > 0; off >>= 1) acc += __shfl_down(acc, off, 32);
  if (lane == 0) out[n] = acc + 256.f * bo[n];
}

// ---------------- launcher ----------------
extern "C" void kernel_launch(void* const* d_in, const int* in_sizes, int n_in,
                              void* d_out, int out_size, void* d_ws, size_t ws_size,
                              hipStream_t stream) {
  (void)in_sizes; (void)n_in; (void)out_size; (void)ws_size;
  const float* sent   = (const float*)d_in[0];
  const float* action = (const float*)d_in[1];
  const int*   pos    = (const int*)d_in[2];
  const float* Wih1   = (const float*)d_in[3];
  const float* Whh1   = (const float*)d_in[4];
  const float* bih1   = (const float*)d_in[5];
  const float* bhh1   = (const float*)d_in[6];
  const float* W1     = (const float*)d_in[7];
  const float* b1     = (const float*)d_in[8];
  const float* Wih2   = (const float*)d_in[9];
  const float* Whh2   = (const float*)d_in[10];
  const float* bih2   = (const float*)d_in[11];
  const float* bhh2   = (const float*)d_in[12];
  const float* W2     = (const float*)d_in[13];
  const float* b2     = (const float*)d_in[14];
  const float* Wi     = (const float*)d_in[15];
  const float* bi     = (const float*)d_in[16];
  const float* Wo     = (const float*)d_in[17];
  const float* bo     = (const float*)d_in[18];

  float* ws   = (float*)d_ws;
  float* out  = (float*)d_out;
  unsigned* sync = (unsigned*)d_ws;
  float* hb1  = ws + OFF_HB1;
  float* hb2  = ws + OFF_HB2;
  float* pre1 = ws + OFF_PRE1;
  float* hall = ws + OFF_HALL;
  float* embg = ws + OFF_EMBG;
  float* emb2 = ws + OFF_EMB2;
  float* pre2 = ws + OFF_PRE2;
  float* ph   = ws + OFF_PH;
  float* sbuf = ws + OFF_S;
  float* qkv  = ws + OFF_QKV;
  float* scor = ws + OFF_SCOR;
  float* aout = ws + OFF_AOUT;
  float* csum = ws + OFF_CSUM;

  // zero barrier counters + h double-buffers (deterministic every call)
  hipMemsetAsync(d_ws, 0, ZERO_FLOATS * sizeof(float), stream);

  // pre1 = sent @ Wih1^T + bih1 + bhh1            (2048 x 2400, K=300)
  gemm16<true, 0><<<dim3(G1 / 16, T_LEN / 16), 32, 0, stream>>>(
      sent, Wih1, bih1, bhh1, pre1, T_LEN, G1, D_IN, D_IN, D_IN, G1, 1.f);

  // LSTM1 scan (persistent, grid-synced, ASYNC h staging)
  lstm1_scan<<<40, 64, 0, stream>>>(pre1, Whh1, action, hb1, hall, sync);

  // gather at action positions
  gather_rows<<<P_LEN, 128, 0, stream>>>(hall, pos, embg);

  // emb2 = tanh(embg @ W1^T + b1)                 (256 x 600, K=600)
  gemm16<true, 1><<<dim3((H1 + 15) / 16, P_LEN / 16), 32, 0, stream>>>(
      embg, W1, b1, nullptr, emb2, P_LEN, H1, H1, H1, H1, H1, 1.f);

  // pre2 = emb2 @ Wih2^T + bih2 + bhh2            (256 x 8192, K=600)
  gemm16<true, 0><<<dim3(G2 / 16, P_LEN / 16), 32, 0, stream>>>(
      emb2, Wih2, bih2, bhh2, pre2, P_LEN, G2, H1, H1, H1, G2, 1.f);

  // LSTM2 scan (persistent, grid-synced, TDM h staging)
  lstm2_scan<<<64, 128, 0, stream>>>(pre2, Whh2, hb2, ph, sync + 4);

  // s = tanh(ph @ W2^T + b2)                      (256 x 2048, K=2048)
  gemm16<true, 1><<<dim3(H2 / 16, P_LEN / 16), 32, 0, stream>>>(
      ph, W2, b2, nullptr, sbuf, P_LEN, H2, H2, H2, H2, H2, 1.f);

  // qkv = s @ in_proj_w^T + in_proj_b             (256 x 6144, K=2048)
  gemm16<true, 0><<<dim3(E3 / 16, P_LEN / 16), 32, 0, stream>>>(
      sbuf, Wi, bi, nullptr, qkv, P_LEN, E3, H2, H2, H2, E3, 1.f);

  // scores = (q @ k^T) / sqrt(E)                  (256 x 256, K=2048)
  gemm16<true, 0><<<dim3(P_LEN / 16, P_LEN / 16), 32, 0, stream>>>(
      qkv, qkv + H2, nullptr, nullptr, scor, P_LEN, P_LEN, H2, E3, E3, P_LEN,
      0.022097086912079608f);

  softmax256<<<P_LEN, 256, 0, stream>>>(scor);

  // aout = attn @ v                               (256 x 2048, K=256, B not transposed)
  gemm16<false, 0><<<dim3(H2 / 16, P_LEN / 16), 32, 0, stream>>>(
      scor, qkv + 2 * H2, nullptr, nullptr, aout, P_LEN, H2, P_LEN, P_LEN, E3, H2, 1.f);

  // final: out = colsum(aout) @ Wo^T + 256*bo
  colsum_k<<<H2 / 256, 256, 0, stream>>>(aout, csum);
  final_matvec<<<H2 / 8, 256, 0, stream>>>(csum, Wo, bo, out);
}